// ProjectLayer_6906307412003
// MI455X (gfx1250) — compile-verified
//
#include <hip/hip_runtime.h>
#include <math.h>

#define N_VIEWS  5
#define N_JOINTS 15
#define HM_H     128
#define HM_W     240
#define HM_SZ    (HM_H * HM_W)              // 30720 floats per (view,joint) map
#define P_NUM    4
#define VDIM     64
#define FXC      253
#define FYC      253
#define FZC      64
#define CUBE_VOX (VDIM * VDIM * VDIM)       // 262144 voxels per (p,joint)
#define CUBES_FLOATS (P_NUM * N_JOINTS * CUBE_VOX)   // 15,728,640
#define GRIDS_FLOATS (3 * 4096 * 2)                  // 24,576

__device__ __forceinline__ int iclamp(int v, int lo, int hi) {
    return v < lo ? lo : (v > hi ? hi : v);
}

// One thread per (p, x, y, z) voxel; z fastest -> coalesced grid loads & stores.
// blockIdx.x in [0, 4096): p = blk >> 10, voxel id = (blk & 1023)*256 + tid.
__global__ __launch_bounds__(256)
void ProjectLayer_cubes_kernel(const float* __restrict__ heat,   // [5][15][128][240]
                               const float* __restrict__ sgrid,  // [5][253][253][64][2]
                               const float* __restrict__ pc,     // [4][7]
                               float* __restrict__ out)          // [4][15][64][64][64]
{
    const int blk = blockIdx.x;
    const int p   = blk >> 10;                               // uniform per block (SALU)
    const int vid = ((blk & 1023) << 8) | threadIdx.x;       // 0..262143
    const int iz  = vid & 63;
    const int iy  = (vid >> 6) & 63;
    const int ix  = vid >> 12;

    // ---- per-proposal scalars (uniform within block) ----
    const float pc0 = pc[p * 7 + 0];
    const float pc1 = pc[p * 7 + 1];
    const float pc2 = pc[p * 7 + 2];
    const float pc5 = pc[p * 7 + 5];
    const float pc6 = pc[p * 7 + 6];

    // mask = max(int((1 - w)/2 * 63), 0); truncating cast, value >= 0
    const int mask_x = max((int)(((1.0f - pc5) * 0.5f) * 63.0f), 0);
    const int mask_y = max((int)(((1.0f - pc6) * 0.5f) * 63.0f), 0);

    // ctl = round((pc + W/2 - C - I/2) / W * (F-1)), round-half-even like jnp.round
    const int cx = (int)rintf((((pc0 + 4000.0f) - 0.0f)    - 1000.0f) / 8000.0f * 252.0f);
    const int cy = (int)rintf((((pc1 + 4000.0f) - (-500.0f)) - 1000.0f) / 8000.0f * 252.0f);
    const int cz = (int)rintf((((pc2 + 1000.0f) - 800.0f)  - 1000.0f) / 2000.0f * 63.0f);

    const int xs = max(-cx, 0) + mask_x;
    const int xe = min(FXC - cx, VDIM) - mask_x;
    const int ys = max(-cy, 0) + mask_y;
    const int ye = min(FYC - cy, VDIM) - mask_y;
    const int zs = max(-cz, 0);
    const int ze = min(FZC - cz, VDIM);

    const bool valid = (ix >= xs) && (ix < xe) &&
                       (iy >= ys) && (iy < ye) &&
                       (iz >= zs) && (iz < ze);

    float acc[N_JOINTS];
#pragma unroll
    for (int j = 0; j < N_JOINTS; ++j) acc[j] = 0.0f;

    if (valid) {
        const int fx = iclamp(cx + ix, 0, FXC - 1);
        const int fy = iclamp(cy + iy, 0, FYC - 1);
        const int fz = iclamp(cz + iz, 0, FZC - 1);
        const int sg_vox = (fx * FYC + fy) * FZC + fz;       // < 2^23, int ok

#pragma unroll
        for (int v = 0; v < N_VIEWS; ++v) {
            // coalesced 8B load: consecutive z -> consecutive fz pairs
            const float2 g = *(const float2*)(sgrid + 2 * (sg_vox + v * (FXC * FYC * FZC)));

            // bilinear setup (shared by all 15 joints of this view)
            const float px = ((g.x + 1.0f) * 0.5f) * (float)(HM_W - 1);
            const float py = ((g.y + 1.0f) * 0.5f) * (float)(HM_H - 1);
            const float fx0 = floorf(px);
            const float fy0 = floorf(py);
            const float wx1 = px - fx0;
            const float wy1 = py - fy0;
            const int ix0 = (int)fx0;
            const int iy0 = (int)fy0;

            const bool okx0 = (ix0 >= 0) && (ix0 < HM_W);
            const bool okx1 = (ix0 + 1 >= 0) && (ix0 + 1 < HM_W);
            const bool oky0 = (iy0 >= 0) && (iy0 < HM_H);
            const bool oky1 = (iy0 + 1 >= 0) && (iy0 + 1 < HM_H);

            const int xc0 = iclamp(ix0, 0, HM_W - 1);
            const int xc1 = iclamp(ix0 + 1, 0, HM_W - 1);
            const int yc0 = iclamp(iy0, 0, HM_H - 1);
            const int yc1 = iclamp(iy0 + 1, 0, HM_H - 1);

            // OOB masks folded into tap weights
            const float w00 = (okx0 && oky0) ? (1.0f - wx1) * (1.0f - wy1) : 0.0f;
            const float w10 = (okx1 && oky0) ? wx1 * (1.0f - wy1)          : 0.0f;
            const float w01 = (okx0 && oky1) ? (1.0f - wx1) * wy1          : 0.0f;
            const float w11 = (okx1 && oky1) ? wx1 * wy1                   : 0.0f;

            const int o00 = yc0 * HM_W + xc0;
            const int o10 = yc0 * HM_W + xc1;
            const int o01 = yc1 * HM_W + xc0;
            const int o11 = yc1 * HM_W + xc1;

            const float* __restrict__ Hv = heat + v * (N_JOINTS * HM_SZ);
#pragma unroll
            for (int j = 0; j < N_JOINTS; ++j) {
                const float* __restrict__ H = Hv + j * HM_SZ;
                float s = H[o00] * w00;                 // 4 gathers / joint
                s = fmaf(H[o10], w10, s);
                s = fmaf(H[o01], w01, s);
                s = fmaf(H[o11], w11, s);
                acc[j] += s;
            }
        }
    }

    // cubes[p][j][x][y][z]; vid == x*4096 + y*64 + z. Streaming NT stores so the
    // 63MB write-only output doesn't evict the L2-resident heatmaps.
    float* __restrict__ obase = out + (size_t)p * (N_JOINTS * CUBE_VOX) + vid;
#pragma unroll
    for (int j = 0; j < N_JOINTS; ++j) {
        float r = acc[j] * 0.2f;                 // mean over 5 views
        r = fminf(fmaxf(r, 0.0f), 1.0f);         // clip(cubes*valid, 0, 1)
        __builtin_nontemporal_store(r, obase + (size_t)j * CUBE_VOX);
    }
}

// Constant "grids" output: 3 planes of 4096 (a,b) pairs from f64 linspace (NumPy
// computes linspace in float64, then the reference casts to f32).
__global__ __launch_bounds__(256)
void ProjectLayer_grids_kernel(float* __restrict__ out)
{
    const int t = blockIdx.x * blockDim.x + threadIdx.x;   // 0..12287
    if (t >= 3 * 4096) return;
    const int s = t >> 12;
    const int r = t & 4095;
    const int a = r >> 6;
    const int b = r & 63;

    const double step = 2000.0 / 63.0;
    const double la = -1000.0 + (double)a * step;
    const double lb = -1000.0 + (double)b * step;

    double v0, v1;
    if (s == 0)      { v0 = la + 0.0;    v1 = lb - 500.0; }  // (gx, gy)
    else if (s == 1) { v0 = la + 0.0;    v1 = lb + 800.0; }  // (gx, gz)
    else             { v0 = la - 500.0;  v1 = lb + 800.0; }  // (gy, gz)

    float* o = out + CUBES_FLOATS + (size_t)t * 2;
    o[0] = (float)v0;
    o[1] = (float)v1;
}

extern "C" void kernel_launch(void* const* d_in, const int* in_sizes, int n_in,
                              void* d_out, int out_size, void* d_ws, size_t ws_size,
                              hipStream_t stream)
{
    (void)in_sizes; (void)n_in; (void)out_size; (void)d_ws; (void)ws_size;

    const float* heat  = (const float*)d_in[0];   // (5,15,128,240) f32
    const float* sgrid = (const float*)d_in[1];   // (5,253,253,64,2) f32
    const float* pc    = (const float*)d_in[2];   // (4,7) f32
    float* out = (float*)d_out;                   // cubes ++ grids, f32

    // 4 proposals * 64^3 voxels, one thread each: 4096 blocks of 256 (8 wave32s)
    ProjectLayer_cubes_kernel<<<4096, 256, 0, stream>>>(heat, sgrid, pc, out);

    // 3*4096 grid pairs
    ProjectLayer_grids_kernel<<<48, 256, 0, stream>>>(out);
}